// MSEDiffRFLoss_40767829573844
// MI455X (gfx1250) — compile-verified
//
#include <hip/hip_runtime.h>
#include <hip/hip_bf16.h>
#include <stdint.h>

// ---------------------------------------------------------------------------
// Problem constants (match the JAX reference)
// ---------------------------------------------------------------------------
#define BB      8
#define HH      512
#define WW      640
#define HW      (HH * WW)           // 327680
#define NPIX    (BB * HW)           // 2621440
#define TOPK    512
#define CAP     24576               // per-image NMS candidate cap (~13k expected)
#define BORD    8
#define GR      7                   // gaussian radius (15x15)
#define FIXSC   16777216.0f         // 2^24 fixed-point scale for deterministic scatter
#define TILE    2048                // pixels per reduction tile (8KB fp32)
#define RGRID   320                 // reduction grid; NPIX/TILE/RGRID == 4 tiles/block
#define RTPB    4

typedef __attribute__((ext_vector_type(2))) float v2f;
typedef __attribute__((ext_vector_type(8))) float v8f;

// ---------------------------------------------------------------------------
// Workspace layout (bytes)
// ---------------------------------------------------------------------------
#define WS_W2      0                            // float[NPIX]   warped+border score2
#define WS_GT      (WS_W2  + NPIX * 4)          // uint32[NPIX]  fixed-point gaussian map
#define WS_CAND    (WS_GT  + NPIX * 4)          // uint64[16*CAP] NMS candidates
#define WS_CNT     (WS_CAND + 16 * CAP * 8)     // int[16] candidate counts (pad 256)
#define WS_SEL     (WS_CNT + 256)               // uint64[16*TOPK] top-k selections
#define WS_PART    (WS_SEL + 16 * TOPK * 8)     // float[RGRID*2] block partials

// ---------------------------------------------------------------------------
// Homography helpers (match reference: p1=(x,y,1), rows of H)
// ---------------------------------------------------------------------------
__device__ __forceinline__ void warp_coords(const float* __restrict__ Hm,
                                            float x, float y,
                                            float& x2, float& y2) {
  float den = Hm[6] * x + Hm[7] * y + Hm[8];
  float inv = 1.0f / den;
  x2 = (Hm[0] * x + Hm[1] * y + Hm[2]) * inv;
  y2 = (Hm[3] * x + Hm[4] * y + Hm[5]) * inv;
}

__device__ __forceinline__ float vis_at(const float* __restrict__ Hm, int x, int y) {
  float x2, y2;
  warp_coords(Hm, (float)x, (float)y, x2, y2);
  float x0 = floorf(x2), y0 = floorf(y2);
  float x1 = x0 + 1.0f, y1 = y0 + 1.0f;
  float wa = (x1 - x2) * (y1 - y2), wb = (x2 - x0) * (y1 - y2);
  float wc = (x1 - x2) * (y2 - y0), wd = (x2 - x0) * (y2 - y0);
  float fa = (x0 >= 0.f && x0 <= (float)(WW - 1) && y0 >= 0.f && y0 <= (float)(HH - 1)) ? 1.f : 0.f;
  float fb = (x1 >= 0.f && x1 <= (float)(WW - 1) && y0 >= 0.f && y0 <= (float)(HH - 1)) ? 1.f : 0.f;
  float fc = (x0 >= 0.f && x0 <= (float)(WW - 1) && y1 >= 0.f && y1 <= (float)(HH - 1)) ? 1.f : 0.f;
  float fd = (x1 >= 0.f && x1 <= (float)(WW - 1) && y1 >= 0.f && y1 <= (float)(HH - 1)) ? 1.f : 0.f;
  float v = wa * fa + wb * fb + wc * fc + wd * fd;
  return v > 0.f ? 1.f : 0.f;
}

// ---------------------------------------------------------------------------
// K1: bilinear homography warp of score2 + border filter -> w2 plane
// ---------------------------------------------------------------------------
__global__ void __launch_bounds__(256) k_warp(const float* __restrict__ score2,
                                              const float* __restrict__ homo,
                                              float* __restrict__ w2plane) {
  int p = blockIdx.x * 256 + threadIdx.x;
  int b = p / HW;
  int rem = p - b * HW;
  int y = rem / WW;
  int x = rem - y * WW;
  const float* Hm = homo + b * 9;
  float x2, y2;
  warp_coords(Hm, (float)x, (float)y, x2, y2);
  float x0 = floorf(x2), y0 = floorf(y2);
  float x1 = x0 + 1.0f, y1 = y0 + 1.0f;
  float wa = (x1 - x2) * (y1 - y2), wb = (x2 - x0) * (y1 - y2);
  float wc = (x1 - x2) * (y2 - y0), wd = (x2 - x0) * (y2 - y0);
  const float* img = score2 + (size_t)b * HW;
  auto samp = [&](float yc, float xc) -> float {
    bool v = (xc >= 0.f) && (xc <= (float)(WW - 1)) && (yc >= 0.f) && (yc <= (float)(HH - 1));
    int xi = (int)fminf(fmaxf(xc, 0.f), (float)(WW - 1));
    int yi = (int)fminf(fmaxf(yc, 0.f), (float)(HH - 1));
    return v ? img[yi * WW + xi] : 0.f;
  };
  float w2 = wa * samp(y0, x0) + wb * samp(y0, x1) + wc * samp(y1, x0) + wd * samp(y1, x1);
  bool bd = (y >= BORD && y < HH - BORD && x >= BORD && x < WW - BORD);
  w2plane[p] = bd ? w2 : 0.f;
}

// ---------------------------------------------------------------------------
// K2: 5x5 NMS + candidate compaction for both maps (which=0: score1*border,
//     which=1: warped plane). Key = (valueBits<<32)|(~idx) -> unique keys so
//     top-k & ordering are deterministic and match lax.top_k tie behavior.
// ---------------------------------------------------------------------------
__global__ void __launch_bounds__(256) k_nms(const float* __restrict__ score1,
                                             const float* __restrict__ w2plane,
                                             uint64_t* __restrict__ cand,
                                             int* __restrict__ cnt) {
  int g = blockIdx.x * 256 + threadIdx.x;
  int which = g / NPIX;
  int p = g - which * NPIX;
  int b = p / HW;
  int rem = p - b * HW;
  int y = rem / WW;
  int x = rem - y * WW;
  auto val = [&](int yy, int xx) -> float {
    if (yy < 0 || yy >= HH || xx < 0 || xx >= WW) return 0.f;
    if (which == 0) {
      bool bd = (yy >= BORD && yy < HH - BORD && xx >= BORD && xx < WW - BORD);
      return bd ? score1[(size_t)b * HW + yy * WW + xx] : 0.f;
    }
    return w2plane[(size_t)b * HW + yy * WW + xx];
  };
  float c = val(y, x);
  if (c <= 0.f) return;                 // NMS_THRESH == 0.0
  bool ismax = true;
  #pragma unroll
  for (int dy = -2; dy <= 2; ++dy)
    #pragma unroll
    for (int dx = -2; dx <= 2; ++dx)
      ismax &= (c >= val(y + dy, x + dx));
  if (!ismax) return;
  int list = which * BB + b;
  int pos = atomicAdd(&cnt[list], 1);
  if (pos < CAP) {
    uint64_t key = ((uint64_t)__float_as_uint(c) << 32) |
                   (uint32_t)(0xFFFFFFFFu - (uint32_t)rem);
    cand[(size_t)list * CAP + pos] = key;
  }
}

// ---------------------------------------------------------------------------
// K3: per-list exact top-512 via 8-pass radix-select over 64-bit keys, then
//     in-LDS bitonic sort (descending output). Lists 0..7 also emit kp1.
// ---------------------------------------------------------------------------
__global__ void __launch_bounds__(256) k_select(const uint64_t* __restrict__ cand,
                                                const int* __restrict__ cnt,
                                                uint64_t* __restrict__ sel,
                                                float* __restrict__ out) {
  int list = blockIdx.x;
  int tid = threadIdx.x;
  const uint64_t* ck = cand + (size_t)list * CAP;
  int n = min(cnt[list], CAP);

  __shared__ int hist[256];
  __shared__ uint64_t sh_prefix;
  __shared__ int sh_k;
  __shared__ int selCnt;
  __shared__ uint64_t skeys[TOPK];

  if (tid == 0) { sh_prefix = 0; sh_k = TOPK; selCnt = 0; }
  __syncthreads();

  uint64_t pmask = 0;
  for (int shift = 56; shift >= 0; shift -= 8) {
    hist[tid] = 0;
    __syncthreads();
    uint64_t prefix = sh_prefix;
    for (int i = tid; i < n; i += 256) {
      uint64_t key = ck[i];
      if ((key & pmask) == prefix)
        atomicAdd(&hist[(int)((key >> shift) & 255)], 1);
    }
    __syncthreads();
    if (tid == 0) {
      int k = sh_k, cum = 0, d = 255;
      for (; d > 0; --d) {
        int h = hist[d];
        if (cum + h >= k) break;
        cum += h;
      }
      sh_k = k - cum;
      sh_prefix = prefix | ((uint64_t)d << shift);
    }
    __syncthreads();
    pmask |= (0xFFull << shift);
  }
  uint64_t T = sh_prefix;   // exact 512th-largest key (keys unique)

  for (int i = tid; i < n; i += 256) {
    uint64_t key = ck[i];
    if (key >= T) {
      int pos = atomicAdd(&selCnt, 1);
      if (pos < TOPK) skeys[pos] = key;
    }
  }
  __syncthreads();
  if (tid == 0) {                       // pad (only if n < 512; never expected)
    for (int c = min(selCnt, TOPK); c < TOPK; ++c) skeys[c] = 0;
  }
  __syncthreads();

  // bitonic sort, ascending; read back reversed for descending rank order
  for (int k = 2; k <= TOPK; k <<= 1) {
    for (int j = k >> 1; j > 0; j >>= 1) {
      for (int i = tid; i < TOPK; i += 256) {
        int ixj = i ^ j;
        if (ixj > i) {
          uint64_t a = skeys[i], bk = skeys[ixj];
          bool asc = ((i & k) == 0);
          if ((a > bk) == asc) { skeys[i] = bk; skeys[ixj] = a; }
        }
      }
      __syncthreads();
    }
  }

  for (int r = tid; r < TOPK; r += 256) {
    uint64_t key = skeys[TOPK - 1 - r];
    sel[(size_t)list * TOPK + r] = key;
    if (list < BB) {                    // kp1 output for score1 lists
      uint32_t idx = 0xFFFFFFFFu - (uint32_t)key;
      int yy = (int)(idx / WW), xx = (int)(idx - (uint32_t)yy * WW);
      float* kp = out + 1 + ((size_t)list * TOPK + r) * 2;
      kp[0] = (float)yy;
      kp[1] = (float)xx;
    }
  }
}

// ---------------------------------------------------------------------------
// K4: scatter gaussian blobs (sigma=0.5, 15x15) from warped top-512 points
//     into fixed-point integer plane (associative -> deterministic).
// ---------------------------------------------------------------------------
__global__ void __launch_bounds__(256) k_blob(const uint64_t* __restrict__ sel,
                                              uint32_t* __restrict__ gtplane) {
  int blk = blockIdx.x;                 // BB*TOPK blocks
  int b = blk / TOPK;
  int r = blk - b * TOPK;
  uint64_t key = sel[(size_t)(BB + b) * TOPK + r];
  float v = __uint_as_float((uint32_t)(key >> 32));
  if (v <= 0.f) return;
  uint32_t idx = 0xFFFFFFFFu - (uint32_t)key;
  int y = (int)(idx / WW), x = (int)(idx - (uint32_t)y * WW);
  int t = threadIdx.x;
  if (t >= 15 * 15) return;
  int dy = t / 15 - GR, dx = t % 15 - GR;
  int yy = y + dy, xx = x + dx;
  if (yy < 0 || yy >= HH || xx < 0 || xx >= WW) return;
  // g1(t)=exp(-t^2/(2*0.25)) -> exp(-2 t^2); separable outer product
  float g = v * expf(-2.0f * (float)(dy * dy + dx * dx));
  uint32_t q = __float2uint_rn(g * FIXSC);
  if (q) atomicAdd(gtplane + (size_t)b * HW + yy * WW + xx, q);
}

// ---------------------------------------------------------------------------
// K5: fused masked-MSE reduction.
//  - Streams score1 and gt tiles HBM->LDS with CDNA5 async global->LDS copies
//    (double buffered; s_wait_asynccnt + barrier before consume).
//  - Recomputes vis mask per pixel (cheap VALU) instead of re-reading a plane.
//  - Feeds (s1-gt)^2*vis and vis through V_WMMA_F32_16X16X4_F32 with an
//    all-ones B matrix: D = A*1 + C accumulates row sums on the matrix pipe,
//    freeing the VALU for the homography math. Wave total = sum(D)/16.
// ---------------------------------------------------------------------------
struct RSMem {            // single __shared__ object -> LDS base offset 0
  float    s1[2][TILE];   // byte off 0      .. 16384
  uint32_t gt[2][TILE];   // byte off 16384  .. 32768
  float    wsumE[8];
  float    wsumN[8];
};

__device__ __forceinline__ void async_ld16(uint32_t ldsOff, const void* g) {
  asm volatile("global_load_async_to_lds_b128 %0, %1, off"
               :: "v"(ldsOff), "v"(g) : "memory");
}
__device__ __forceinline__ void wait_async0() {
  asm volatile("s_wait_asynccnt 0" ::: "memory");
}

__global__ void __launch_bounds__(256) k_reduce(const float* __restrict__ score1,
                                                const uint32_t* __restrict__ gtplane,
                                                const float* __restrict__ homo,
                                                float* __restrict__ partials) {
  __shared__ RSMem sm;
  const int tid = threadIdx.x;

  auto issue = [&](int buf, int tileStart) {
    const char* g1 = (const char*)(score1 + tileStart);
    const char* g2 = (const char*)(gtplane + tileStart);
    uint32_t o1 = (uint32_t)(buf * TILE * 4);          // sm.s1[buf] LDS byte off
    uint32_t o2 = (uint32_t)(2 * TILE * 4 + buf * TILE * 4); // sm.gt[buf]
    uint32_t o = (uint32_t)tid * 16;
    async_ld16(o1 + o,        g1 + o);
    async_ld16(o1 + o + 4096, g1 + o + 4096);
    async_ld16(o2 + o,        g2 + o);
    async_ld16(o2 + o + 4096, g2 + o + 4096);
  };

  v8f accE = {};
  v8f accN = {};
  const v2f ones2 = {1.0f, 1.0f};

  int t0 = blockIdx.x * RTPB;           // 4 contiguous tiles per block
  int cur = 0;
  issue(cur, t0 * TILE);

  for (int i = 0; i < RTPB; ++i) {
    int tileStart = (t0 + i) * TILE;
    wait_async0();                      // my async copies into buf[cur] done
    __syncthreads();                    // everyone's copies visible
    if (i + 1 < RTPB) issue(cur ^ 1, (t0 + i + 1) * TILE);

    int b = tileStart / HW;             // tiles never span images (HW%TILE==0)
    const float* Hm = homo + b * 9;

    #pragma unroll
    for (int k = 0; k < 4; ++k) {
      float e[2], vv[2];
      #pragma unroll
      for (int u = 0; u < 2; ++u) {
        int j = tid + (2 * k + u) * 256;
        int p = tileStart + j;
        int rem = p - b * HW;
        int y = rem / WW;
        int x = rem - y * WW;
        float s1v = sm.s1[cur][j];
        bool bd = (y >= BORD && y < HH - BORD && x >= BORD && x < WW - BORD);
        s1v = bd ? s1v : 0.f;
        float gt = (float)sm.gt[cur][j] * (1.0f / FIXSC);
        float vis = vis_at(Hm, x, y);
        float d = s1v - gt;
        e[u] = d * d * vis;
        vv[u] = vis;
      }
      v2f ae = {e[0], e[1]};
      v2f an = {vv[0], vv[1]};
      // D = A x ones(4x16) + C : matrix-pipe accumulation of the reduction
      accE = __builtin_amdgcn_wmma_f32_16x16x4_f32(false, ae, false, ones2,
                                                   (short)0, accE, false, false);
      accN = __builtin_amdgcn_wmma_f32_16x16x4_f32(false, an, false, ones2,
                                                   (short)0, accN, false, false);
    }
    __syncthreads();
    cur ^= 1;
  }

  // Each row-sum is replicated across the 16 columns of D -> total = sum/16.
  float le = 0.f, ln = 0.f;
  #pragma unroll
  for (int i = 0; i < 8; ++i) { le += accE[i]; ln += accN[i]; }
  for (int off = 16; off > 0; off >>= 1) {
    le += __shfl_down(le, off, 32);
    ln += __shfl_down(ln, off, 32);
  }
  int wave = tid >> 5, lane = tid & 31;
  if (lane == 0) { sm.wsumE[wave] = le; sm.wsumN[wave] = ln; }
  __syncthreads();
  if (tid == 0) {
    float te = 0.f, tn = 0.f;
    for (int w = 0; w < 8; ++w) { te += sm.wsumE[w]; tn += sm.wsumN[w]; }
    partials[blockIdx.x * 2 + 0] = te * (1.0f / 16.0f);
    partials[blockIdx.x * 2 + 1] = tn * (1.0f / 16.0f);
  }
}

// ---------------------------------------------------------------------------
// K6: deterministic serial reduction of block partials -> loss
// ---------------------------------------------------------------------------
__global__ void k_final(const float* __restrict__ partials, float* __restrict__ out) {
  __shared__ float se[RGRID], sn[RGRID];
  int tid = threadIdx.x;
  if (tid < RGRID) { se[tid] = partials[tid * 2]; sn[tid] = partials[tid * 2 + 1]; }
  __syncthreads();
  if (tid == 0) {
    float e = 0.f, n = 0.f;
    for (int i = 0; i < RGRID; ++i) { e += se[i]; n += sn[i]; }
    out[0] = e / n * 100.0f;           // LOSS_LAMBDA
  }
}

// ---------------------------------------------------------------------------
extern "C" void kernel_launch(void* const* d_in, const int* in_sizes, int n_in,
                              void* d_out, int out_size, void* d_ws, size_t ws_size,
                              hipStream_t stream) {
  (void)in_sizes; (void)n_in; (void)out_size; (void)ws_size;
  const float* score1 = (const float*)d_in[0];
  const float* score2 = (const float*)d_in[1];
  const float* homo   = (const float*)d_in[2];
  float* out = (float*)d_out;

  char* ws = (char*)d_ws;
  float*    w2plane  = (float*)(ws + WS_W2);
  uint32_t* gtplane  = (uint32_t*)(ws + WS_GT);
  uint64_t* cand     = (uint64_t*)(ws + WS_CAND);
  int*      cnt      = (int*)(ws + WS_CNT);
  uint64_t* sel      = (uint64_t*)(ws + WS_SEL);
  float*    partials = (float*)(ws + WS_PART);

  // per-call re-init (graph-capture safe)
  hipMemsetAsync(gtplane, 0, (size_t)NPIX * 4, stream);
  hipMemsetAsync(cnt, 0, 16 * sizeof(int), stream);

  k_warp  <<<NPIX / 256,     256, 0, stream>>>(score2, homo, w2plane);
  k_nms   <<<2 * NPIX / 256, 256, 0, stream>>>(score1, w2plane, cand, cnt);
  k_select<<<16,             256, 0, stream>>>(cand, cnt, sel, out);
  k_blob  <<<BB * TOPK,      256, 0, stream>>>(sel, gtplane);
  k_reduce<<<RGRID,          256, 0, stream>>>(score1, gtplane, homo, partials);
  k_final <<<1,              RGRID, 0, stream>>>(partials, out);
}